// GCN_33337536151584
// MI455X (gfx1250) — compile-verified
//
#include <hip/hip_runtime.h>
#include <math.h>

#define N_NODES 100000
#define N_EDGES 1600000
#define NFEAT   128
#define NHID    64
#define NCLASS  16
#define ROWTILES (N_NODES / 16)   // 6250, exact

typedef float v2f __attribute__((ext_vector_type(2)));
typedef float v8f __attribute__((ext_vector_type(8)));

// ---------------- utility ----------------
__global__ void zero_kernel(float* __restrict__ p, int n) {
    int i = blockIdx.x * blockDim.x + threadIdx.x;
    if (i < n) p[i] = 0.0f;
}

__global__ void degree_kernel(const int* __restrict__ src, const int* __restrict__ dst,
                              float* __restrict__ outdeg, float* __restrict__ indeg) {
    int e = blockIdx.x * blockDim.x + threadIdx.x;
    if (e < N_EDGES) {
        atomicAdd(&outdeg[src[e]], 1.0f);
        atomicAdd(&indeg[dst[e]], 1.0f);
    }
}

// deg -> 1/sqrt(max(deg,1)), in place (applied to both deg arrays at once)
__global__ void norm_kernel(float* __restrict__ deg, int n) {
    int i = blockIdx.x * blockDim.x + threadIdx.x;
    if (i < n) {
        float d = deg[i];
        d = d < 1.0f ? 1.0f : d;
        deg[i] = 1.0f / sqrtf(d);
    }
}

// ---------------- GEMM1: h1 = (x @ W1) * norm_src[:,None]  (M=100000,K=128,N=64) ----
// block = 128 threads = 4 waves; block b covers rows [16b,16b+16); wave w covers cols [16w,16w+16)
__global__ __launch_bounds__(128) void gemm1_wmma(const float* __restrict__ x,
                                                  const float* __restrict__ W1,
                                                  const float* __restrict__ norm_src,
                                                  float* __restrict__ h1) {
    const int wave  = threadIdx.x >> 5;
    const int lane  = threadIdx.x & 31;
    const int row0  = blockIdx.x * 16;
    const int n0    = wave * 16;
    const int m     = lane & 15;          // A row within tile
    const int n     = lane & 15;          // B/D col within tile
    const int khalf = (lane >> 4) * 2;    // 0 for lanes 0-15, 2 for lanes 16-31
    const int hi8   = (lane >> 4) * 8;    // C/D row offset

    const float* __restrict__ xrow = x + (size_t)(row0 + m) * NFEAT;
    v8f c = {};
#pragma unroll
    for (int k = 0; k < NFEAT; k += 4) {
        v2f a, b;
        a.x = xrow[k + khalf + 0];
        a.y = xrow[k + khalf + 1];
        b.x = W1[(size_t)(k + khalf + 0) * NHID + n0 + n];
        b.y = W1[(size_t)(k + khalf + 1) * NHID + n0 + n];
        c = __builtin_amdgcn_wmma_f32_16x16x4_f32(false, a, false, b,
                                                  (short)0, c, false, false);
    }
#pragma unroll
    for (int v = 0; v < 8; ++v) {
        int row = row0 + v + hi8;
        h1[(size_t)row * NHID + n0 + n] = c[v] * norm_src[row];
    }
}

// ---------------- GEMM2: h2 = (h @ W2) * norm_src[:,None]  (M=100000,K=64,N=16) -----
// block = 128 threads = 4 waves; each wave handles one 16-row tile (whole N=16)
__global__ __launch_bounds__(128) void gemm2_wmma(const float* __restrict__ h,
                                                  const float* __restrict__ W2,
                                                  const float* __restrict__ norm_src,
                                                  float* __restrict__ h2) {
    const int wave    = threadIdx.x >> 5;
    const int lane    = threadIdx.x & 31;
    const int rowTile = blockIdx.x * 4 + wave;
    if (rowTile >= ROWTILES) return;      // wave-uniform: whole wave skips together
    const int row0  = rowTile * 16;
    const int m     = lane & 15;
    const int n     = lane & 15;
    const int khalf = (lane >> 4) * 2;
    const int hi8   = (lane >> 4) * 8;

    const float* __restrict__ hrow = h + (size_t)(row0 + m) * NHID;
    v8f c = {};
#pragma unroll
    for (int k = 0; k < NHID; k += 4) {
        v2f a, b;
        a.x = hrow[k + khalf + 0];
        a.y = hrow[k + khalf + 1];
        b.x = W2[(size_t)(k + khalf + 0) * NCLASS + n];
        b.y = W2[(size_t)(k + khalf + 1) * NCLASS + n];
        c = __builtin_amdgcn_wmma_f32_16x16x4_f32(false, a, false, b,
                                                  (short)0, c, false, false);
    }
#pragma unroll
    for (int v = 0; v < 8; ++v) {
        int row = row0 + v + hi8;
        h2[(size_t)row * NCLASS + n] = c[v] * norm_src[row];
    }
}

// ---------------- edge scatter: agg[dst] += h[src], F floats per node ---------------
// 16 threads/edge, 4 floats each (F=64)
__global__ void scatter64_kernel(const float* __restrict__ h, const int* __restrict__ src,
                                 const int* __restrict__ dst, float* __restrict__ agg) {
    long long tid = (long long)blockIdx.x * blockDim.x + threadIdx.x;
    int e  = (int)(tid >> 4);
    int f  = (int)(tid & 15) * 4;
    if (e >= N_EDGES) return;
    int s = src[e], d = dst[e];
    float4 v = *(const float4*)(h + (size_t)s * NHID + f);
    float* ap = agg + (size_t)d * NHID + f;
    atomicAdd(ap + 0, v.x); atomicAdd(ap + 1, v.y);
    atomicAdd(ap + 2, v.z); atomicAdd(ap + 3, v.w);
}

// 4 threads/edge, 4 floats each (F=16)
__global__ void scatter16_kernel(const float* __restrict__ h, const int* __restrict__ src,
                                 const int* __restrict__ dst, float* __restrict__ agg) {
    long long tid = (long long)blockIdx.x * blockDim.x + threadIdx.x;
    int e  = (int)(tid >> 2);
    int f  = (int)(tid & 3) * 4;
    if (e >= N_EDGES) return;
    int s = src[e], d = dst[e];
    float4 v = *(const float4*)(h + (size_t)s * NCLASS + f);
    float* ap = agg + (size_t)d * NCLASS + f;
    atomicAdd(ap + 0, v.x); atomicAdd(ap + 1, v.y);
    atomicAdd(ap + 2, v.z); atomicAdd(ap + 3, v.w);
}

// ---------------- layer-1 epilogue: agg = relu(agg*norm_dst + b1), in place ---------
__global__ void post1_kernel(float* __restrict__ agg, const float* __restrict__ norm_dst,
                             const float* __restrict__ b1) {
    int i = blockIdx.x * blockDim.x + threadIdx.x;
    if (i < N_NODES * NHID) {
        int row = i / NHID, col = i - row * NHID;
        float v = agg[i] * norm_dst[row] + b1[col];
        agg[i] = v > 0.0f ? v : 0.0f;
    }
}

// ---------------- final: out = log_softmax(agg2*norm_dst + b2) ----------------------
__global__ void final_kernel(const float* __restrict__ agg2, const float* __restrict__ norm_dst,
                             const float* __restrict__ b2, float* __restrict__ out) {
    int i = blockIdx.x * blockDim.x + threadIdx.x;
    if (i >= N_NODES) return;
    float nd = norm_dst[i];
    float v[NCLASS];
    float mx = -INFINITY;
#pragma unroll
    for (int c = 0; c < NCLASS; ++c) {
        v[c] = agg2[(size_t)i * NCLASS + c] * nd + b2[c];
        mx = fmaxf(mx, v[c]);
    }
    float s = 0.0f;
#pragma unroll
    for (int c = 0; c < NCLASS; ++c) s += expf(v[c] - mx);
    float lse = mx + logf(s);
#pragma unroll
    for (int c = 0; c < NCLASS; ++c) out[(size_t)i * NCLASS + c] = v[c] - lse;
}

extern "C" void kernel_launch(void* const* d_in, const int* in_sizes, int n_in,
                              void* d_out, int out_size, void* d_ws, size_t ws_size,
                              hipStream_t stream) {
    const float* x   = (const float*)d_in[0];   // 100000 x 128
    const int*   src = (const int*)  d_in[1];   // 1.6M
    const int*   dst = (const int*)  d_in[2];   // 1.6M
    const float* W1  = (const float*)d_in[3];   // 128 x 64
    const float* b1  = (const float*)d_in[4];   // 64
    const float* W2  = (const float*)d_in[5];   // 64 x 16
    const float* b2  = (const float*)d_in[6];   // 16
    float* out = (float*)d_out;                 // 100000 x 16

    // workspace layout (floats): total 16.2M floats = 64.8 MB
    float* ws       = (float*)d_ws;
    float* norm_src = ws;                    // deg_out -> norm_src (100000)
    float* norm_dst = ws + 100000;           // deg_in  -> norm_dst (100000)
    float* h1       = ws + 200000;           // 6,400,000
    float* agg1     = h1   + (size_t)N_NODES * NHID;    // 6,400,000
    float* h2       = agg1 + (size_t)N_NODES * NHID;    // 1,600,000
    float* agg2     = h2   + (size_t)N_NODES * NCLASS;  // 1,600,000

    const int T = 256;

    // zero accumulators (must happen every call: harness doesn't re-poison)
    zero_kernel<<<(200000 + T - 1) / T, T, 0, stream>>>(norm_src, 200000);
    zero_kernel<<<((N_NODES * NHID) + T - 1) / T, T, 0, stream>>>(agg1, N_NODES * NHID);
    zero_kernel<<<((N_NODES * NCLASS) + T - 1) / T, T, 0, stream>>>(agg2, N_NODES * NCLASS);

    // degrees -> norms (both arrays contiguous, one norm pass)
    degree_kernel<<<(N_EDGES + T - 1) / T, T, 0, stream>>>(src, dst, norm_src, norm_dst);
    norm_kernel<<<(200000 + T - 1) / T, T, 0, stream>>>(norm_src, 200000);

    // layer 1
    gemm1_wmma<<<ROWTILES, 128, 0, stream>>>(x, W1, norm_src, h1);
    {
        long long threads = (long long)N_EDGES * 16;
        scatter64_kernel<<<(int)((threads + T - 1) / T), T, 0, stream>>>(h1, src, dst, agg1);
    }
    post1_kernel<<<((N_NODES * NHID) + T - 1) / T, T, 0, stream>>>(agg1, norm_dst, b1);

    // layer 2
    gemm2_wmma<<<(ROWTILES + 3) / 4, 128, 0, stream>>>(agg1, W2, norm_src, h2);
    {
        long long threads = (long long)N_EDGES * 4;
        scatter16_kernel<<<(int)((threads + T - 1) / T), T, 0, stream>>>(h2, src, dst, agg2);
    }
    final_kernel<<<(N_NODES + T - 1) / T, T, 0, stream>>>(agg2, norm_dst, b2, out);
}